// TransposedAttention_21406117003505
// MI455X (gfx1250) — compile-verified
//
#include <hip/hip_runtime.h>
#include <hip/hip_bf16.h>

// ---------------------------------------------------------------------------
// Transposed (channel) attention, f32, MI455X / gfx1250, wave32 + WMMA f32.
//
// Fusion plan (memory-bound at 23.3 TB/s; all GEMMs via v_wmma_f32_16x16x4_f32):
//   K1: x --(WMMA qkv GEMM + bias)--> LDS --(dw 3x3)--> dwqkv  [+ q/k norm atomics]
//   K2: finalize 1/max(||row||,eps)
//   K3: sim = q.k^T   (split-K WMMA, f32 atomics)
//   K4: softmax48 with invq*invk*temperature folded in
//   K5: M2 = proj_w @ blockdiag(attn)    (skips materializing attn@v)
//   K6: out = M2 @ v + proj_b            (WMMA)
// ---------------------------------------------------------------------------

typedef float v2f __attribute__((ext_vector_type(2)));
typedef float v8f __attribute__((ext_vector_type(8)));

#define HW    16384   // 128*128
#define CDIM  192
#define ODIM  576
#define BATCH 8
#define NHEAD 4
#define HCH   48
#define YT    8       // output rows per fused tile

__device__ __forceinline__ v8f wmma4(v2f a, v2f b, v8f c) {
    // V_WMMA_F32_16X16X4_F32 : D = A(16x4) * B(4x16) + C(16x16), f32
    return __builtin_amdgcn_wmma_f32_16x16x4_f32(
        false, a, false, b, (short)0, c, false, false);
}

// ---------------------------------------------------------------------------
// Kernel 1 (fused): qkv 1x1-conv GEMM + bias -> LDS -> depthwise 3x3 -> dwqkv,
// with per-channel sum-of-squares atomics for the q/k L2 norms.
// grid = (128/YT, ODIM/16, BATCH), block = 320 (10 waves)
// Wave w computes input image row y0-1+w for 16 channels via WMMA.
// ---------------------------------------------------------------------------
__global__ __launch_bounds__(320)
void qkv_dw_fused(const float* __restrict__ x,   const float* __restrict__ w,
                  const float* __restrict__ bias, const float* __restrict__ dww,
                  float* __restrict__ dwqkv, float* __restrict__ ssq)
{
    __shared__ float lds[10 * 16 * 128];   // 80 KB: [inrow][ch_local][xx]
    const int tid  = threadIdx.x;
    const int lane = tid & 31;
    const int wid  = tid >> 5;             // 0..9  -> input row y0-1+wid
    const int m    = lane & 15;
    const int hi   = lane >> 4;
    const int y0    = blockIdx.x * YT;
    const int obase = blockIdx.y * 16;
    const int b     = blockIdx.z;

    const int y = y0 - 1 + wid;
    if ((unsigned)y < 128u) {
        const float* X = x + (size_t)b * CDIM * HW + y * 128;
        v8f acc[8] = {};
        for (int k = 0; k < CDIM; k += 4) {
            const int c0 = k + 2 * hi;
            v2f a;
            a.x = w[(obase + m) * CDIM + c0];
            a.y = w[(obase + m) * CDIM + c0 + 1];
            const float* r0 = X + (size_t)c0 * HW + m;
            const float* r1 = r0 + HW;
#pragma unroll
            for (int t = 0; t < 8; ++t) {
                v2f bf; bf.x = r0[t * 16]; bf.y = r1[t * 16];
                acc[t] = wmma4(a, bf, acc[t]);
            }
        }
#pragma unroll
        for (int t = 0; t < 8; ++t) {
            const int xx = t * 16 + m;
#pragma unroll
            for (int r = 0; r < 8; ++r) {
                const int ol = r + 8 * hi;
                lds[(wid * 16 + ol) * 128 + xx] = acc[t][r] + bias[obase + ol];
            }
        }
    } else {
        for (int i = lane; i < 16 * 128; i += 32) lds[wid * 16 * 128 + i] = 0.f;
    }
    __syncthreads();

    // depthwise 3x3 over the 8 interior rows, straight out of LDS
    const int which = (obase < CDIM) ? 0 : ((obase < 2 * CDIM) ? 1 : 2); // q,k,v
    float* outp = dwqkv + ((size_t)b * ODIM + obase) * HW;
#pragma unroll 1
    for (int ol = 0; ol < 16; ++ol) {
        const int chg = obase + ol;
        const float* wk = dww + chg * 9;
        const float w00 = wk[0], w01 = wk[1], w02 = wk[2];
        const float w10 = wk[3], w11 = wk[4], w12 = wk[5];
        const float w20 = wk[6], w21 = wk[7], w22 = wk[8];
        float ss = 0.f;
        for (int i = tid; i < YT * 128; i += 320) {
            const int ry = i >> 7;          // 0..7
            const int xx = i & 127;
            const float* L0 = &lds[((ry + 0) * 16 + ol) * 128];
            const float* L1 = L0 + 16 * 128;
            const float* L2 = L1 + 16 * 128;
            float s = w01 * L0[xx] + w11 * L1[xx] + w21 * L2[xx];
            if (xx > 0)   s += w00 * L0[xx - 1] + w10 * L1[xx - 1] + w20 * L2[xx - 1];
            if (xx < 127) s += w02 * L0[xx + 1] + w12 * L1[xx + 1] + w22 * L2[xx + 1];
            outp[(size_t)ol * HW + (y0 + ry) * 128 + xx] = s;
            ss += s * s;
        }
        if (which < 2) {
            // wave32 reduce then one atomic per wave per channel
#pragma unroll
            for (int off = 16; off > 0; off >>= 1)
                ss += __shfl_xor(ss, off, 32);
            if (lane == 0)
                atomicAdd(&ssq[which * 1536 + b * CDIM + (chg - which * CDIM)], ss);
        }
    }
}

// ---------------------------------------------------------------------------
// Kernel 2: invn = 1 / max(sqrt(ssq), eps)   (3072 entries)
// ---------------------------------------------------------------------------
__global__ __launch_bounds__(256)
void finalize_invn(const float* __restrict__ ssq, float* __restrict__ invn)
{
    const int i = blockIdx.x * 256 + threadIdx.x;
    if (i < 3072) invn[i] = 1.0f / fmaxf(sqrtf(ssq[i]), 1e-12f);
}

// ---------------------------------------------------------------------------
// zero fill
// ---------------------------------------------------------------------------
__global__ __launch_bounds__(256)
void zerof(float* __restrict__ p, int n)
{
    const int i = blockIdx.x * 256 + threadIdx.x;
    if (i < n) p[i] = 0.f;
}

// ---------------------------------------------------------------------------
// Kernel 3: split-K GEMM  sim[bh, cq, ck] += sum_n q[cq,n] * k[ck,n]
// One wave computes a full 48x48 tile over a 256-wide K chunk; atomicAdd out.
// grid = (64 chunks, 32 bh), block = 32
// ---------------------------------------------------------------------------
__global__ __launch_bounds__(32)
void sim_gemm(const float* __restrict__ dwqkv, float* __restrict__ sim)
{
    const int lane = threadIdx.x;
    const int m  = lane & 15;
    const int hi = lane >> 4;
    const int n0 = blockIdx.x * 256;
    const int bh = blockIdx.y;                 // b*4 + head
    const int b = bh >> 2, head = bh & 3;

    const float* q    = dwqkv + ((size_t)b * ODIM + head * HCH) * HW;
    const float* kmat = q + (size_t)CDIM * HW;

    v8f acc[3][3] = {};
    for (int k = 0; k < 256; k += 4) {
        const int n = n0 + k + 2 * hi;
        v2f a[3], bf[3];
#pragma unroll
        for (int i = 0; i < 3; ++i) {
            const float* qr = q    + (size_t)(i * 16 + m) * HW + n;
            const float* kr = kmat + (size_t)(i * 16 + m) * HW + n;
            a[i].x  = qr[0]; a[i].y  = qr[1];
            bf[i].x = kr[0]; bf[i].y = kr[1];
        }
#pragma unroll
        for (int i = 0; i < 3; ++i)
#pragma unroll
            for (int j = 0; j < 3; ++j)
                acc[i][j] = wmma4(a[i], bf[j], acc[i][j]);
    }
    float* S = sim + (size_t)bh * HCH * HCH;
#pragma unroll
    for (int i = 0; i < 3; ++i)
#pragma unroll
        for (int j = 0; j < 3; ++j)
#pragma unroll
            for (int r = 0; r < 8; ++r) {
                const int cq = i * 16 + r + 8 * hi;
                const int ck = j * 16 + m;
                atomicAdd(&S[cq * HCH + ck], acc[i][j][r]);
            }
}

// ---------------------------------------------------------------------------
// Kernel 4: fold in 1/||q||, 1/||k||, temperature; softmax over ck (48).
// grid = 8*4*48 rows, block = 64
// ---------------------------------------------------------------------------
__global__ __launch_bounds__(64)
void softmax48(const float* __restrict__ sim, const float* __restrict__ invn,
               const float* __restrict__ temp, float* __restrict__ attn)
{
    const int row = blockIdx.x;           // bh*48 + cq
    const int bh = row / HCH, cq = row % HCH;
    const int b = bh >> 2, head = bh & 3;
    const int t = threadIdx.x;

    const float* S = sim + (size_t)bh * HCH * HCH + cq * HCH;
    const float iq = invn[b * CDIM + head * HCH + cq];
    const float T  = temp[head];

    __shared__ float buf[64];
    float val = 0.f, v = -3.4e38f;
    if (t < HCH) {
        const float ik = invn[1536 + b * CDIM + head * HCH + t];
        val = S[t] * iq * ik * T;
        v = val;
    }
    buf[t] = v; __syncthreads();
    for (int off = 32; off > 0; off >>= 1) {
        if (t < off) buf[t] = fmaxf(buf[t], buf[t + off]);
        __syncthreads();
    }
    const float mx = buf[0]; __syncthreads();
    const float e = (t < HCH) ? expf(val - mx) : 0.f;
    buf[t] = e; __syncthreads();
    for (int off = 32; off > 0; off >>= 1) {
        if (t < off) buf[t] += buf[t + off];
        __syncthreads();
    }
    const float s = buf[0];
    if (t < HCH) attn[(size_t)bh * HCH * HCH + cq * HCH + t] = e / s;
}

// ---------------------------------------------------------------------------
// Kernel 5: M2[b][o][head*48+d] = sum_ch proj_w[o, head*48+ch] * attn[b,head,ch,d]
// ---------------------------------------------------------------------------
__global__ __launch_bounds__(256)
void fuse_proj_attn(const float* __restrict__ pw, const float* __restrict__ attn,
                    float* __restrict__ m2)
{
    const size_t idx = (size_t)blockIdx.x * 256 + threadIdx.x;
    if (idx >= (size_t)BATCH * CDIM * CDIM) return;
    const int c = (int)(idx % CDIM);
    const int o = (int)((idx / CDIM) % CDIM);
    const int b = (int)(idx / (CDIM * CDIM));
    const int head = c / HCH, d = c % HCH;
    const float* A = attn + ((size_t)(b * NHEAD + head)) * HCH * HCH + d;
    const float* W = pw + o * CDIM + head * HCH;
    float s = 0.f;
#pragma unroll
    for (int ch = 0; ch < HCH; ++ch) s += W[ch] * A[ch * HCH];
    m2[idx] = s;
}

// ---------------------------------------------------------------------------
// Kernel 6: out[b,o,n] = sum_c M2[b][o,c] * V[b,c,n] + proj_b[o]
// grid = (HW/128, 192/16, 8), block = 32
// ---------------------------------------------------------------------------
__global__ __launch_bounds__(32)
void out_gemm(const float* __restrict__ m2, const float* __restrict__ dwqkv,
              const float* __restrict__ bias, float* __restrict__ out)
{
    const int lane = threadIdx.x;
    const int m  = lane & 15;
    const int hi = lane >> 4;
    const int nbase = blockIdx.x * 128;
    const int obase = blockIdx.y * 16;
    const int b     = blockIdx.z;

    const float* M = m2 + (size_t)b * CDIM * CDIM;
    const float* V = dwqkv + ((size_t)b * ODIM + 2 * CDIM) * HW;

    v8f acc[8] = {};
    for (int k = 0; k < CDIM; k += 4) {
        const int c0 = k + 2 * hi;
        v2f a;
        a.x = M[(obase + m) * CDIM + c0];
        a.y = M[(obase + m) * CDIM + c0 + 1];
        const float* r0 = V + (size_t)c0 * HW + nbase + m;
        const float* r1 = r0 + HW;
#pragma unroll
        for (int t = 0; t < 8; ++t) {
            v2f bf; bf.x = r0[t * 16]; bf.y = r1[t * 16];
            acc[t] = wmma4(a, bf, acc[t]);
        }
    }
#pragma unroll
    for (int t = 0; t < 8; ++t) {
        const int n = nbase + t * 16 + m;
#pragma unroll
        for (int r = 0; r < 8; ++r) {
            const int o = obase + r + 8 * hi;
            out[((size_t)b * CDIM + o) * HW + n] = acc[t][r] + bias[o];
        }
    }
}

// ---------------------------------------------------------------------------
// Host-side launch
// ---------------------------------------------------------------------------
extern "C" void kernel_launch(void* const* d_in, const int* in_sizes, int n_in,
                              void* d_out, int out_size, void* d_ws, size_t ws_size,
                              hipStream_t stream)
{
    const float* x      = (const float*)d_in[0];
    const float* qkv_w  = (const float*)d_in[1];
    const float* qkv_b  = (const float*)d_in[2];
    const float* dw_w   = (const float*)d_in[3];
    const float* proj_w = (const float*)d_in[4];
    const float* proj_b = (const float*)d_in[5];
    const float* temp   = (const float*)d_in[6];
    float* out = (float*)d_out;

    // workspace layout (floats) -- ssq and sim adjacent so one zero-fill covers both
    float* ws = (float*)d_ws;
    const size_t DW_N = (size_t)BATCH * ODIM * HW;    // 75,497,472
    float* dwqkv = ws;
    float* ssq   = dwqkv + DW_N;                      // 3072
    float* sim   = ssq + 3072;                        // 32*48*48 = 73728
    float* invn  = sim + 73728;                       // 3072
    float* attn  = invn + 3072;                       // 73728
    float* m2    = attn + 73728;                      // 8*192*192 = 294912
    (void)ws_size; (void)n_in; (void)in_sizes; (void)out_size;

    // 0) zero ssq + sim (contiguous 76800 floats)
    zerof<<<(76800 + 255) / 256, 256, 0, stream>>>(ssq, 76800);

    // 1) fused: qkv GEMM (WMMA) + bias + depthwise 3x3 + q/k sum-of-squares
    qkv_dw_fused<<<dim3(128 / YT, ODIM / 16, BATCH), 320, 0, stream>>>(
        x, qkv_w, qkv_b, dw_w, dwqkv, ssq);

    // 2) finalize inverse norms
    finalize_invn<<<(3072 + 255) / 256, 256, 0, stream>>>(ssq, invn);

    // 3) sim = q @ k^T  (split-K WMMA + f32 atomics)
    sim_gemm<<<dim3(64, BATCH * NHEAD), 32, 0, stream>>>(dwqkv, sim);

    // 4) scale + softmax over 48
    softmax48<<<BATCH * NHEAD * HCH, 64, 0, stream>>>(sim, invn, temp, attn);

    // 5) M2 = proj_w @ blockdiag(attn)
    fuse_proj_attn<<<(BATCH * CDIM * CDIM + 255) / 256, 256, 0, stream>>>(proj_w, attn, m2);

    // 6) out = M2 @ v + proj_b  (WMMA, fused projection)
    out_gemm<<<dim3(HW / 128, CDIM / 16, BATCH), 32, 0, stream>>>(m2, dwqkv, proj_b, out);
}